// GraphAttentionNetwork_85856396247183
// MI455X (gfx1250) — compile-verified
//
#include <hip/hip_runtime.h>
#include <hip/hip_bf16.h>

#define N_NODES 50000
#define N_EDGES 1600000
#define N_ETOT  (N_EDGES + N_NODES)
#define HID     128
#define HEADS   4

typedef __attribute__((ext_vector_type(2))) float v2f;
typedef __attribute__((ext_vector_type(8))) float v8f;

// ---- order-preserving float<->uint encoding for atomicMax on floats ----
__device__ __forceinline__ unsigned f2ord(float f) {
    unsigned u = __float_as_uint(f);
    return (u & 0x80000000u) ? ~u : (u | 0x80000000u);
}
__device__ __forceinline__ float ord2f(unsigned u) {
    return (u & 0x80000000u) ? __uint_as_float(u & 0x7fffffffu)
                             : __uint_as_float(~u);
}
// f2ord(-inf) = ~0xFF800000 = 0x007FFFFF
#define ORD_NEG_INF 0x007FFFFFu

// =====================================================================
// fp32 WMMA GEMM: C[M,N] = act(A[M,K] @ B[K,N] + bias)
// One wave computes a 16x64 C tile: 4 accumulators share one A fragment,
// so each K-step issues 1 A load + 4 B-fragment loads + 4 WMMAs.
// V_WMMA_F32_16X16X4_F32 fragment layouts:
//   A (16x4): lanes 0-15 rows, VGPR0/1 = K{0,1} (lanes<16) or K{2,3}.
//   B (4x16): VGPR0 = row K0 (lanes 0-15) / K2 (lanes 16-31), VGPR1 = K1/K3.
//   C: VGPR j -> (row = j + 8*lane[4], col = lane[3:0]).
// Requires: M % 16 == 0, N % 64 == 0, K % 4 == 0.
// =====================================================================
__global__ void gemm_wmma_f32(const float* __restrict__ A, const float* __restrict__ B,
                              const float* __restrict__ bias, float* __restrict__ C,
                              int M, int N, int K, int relu)
{
    const int lane = threadIdx.x & 31;
    const int wid  = blockIdx.x * (blockDim.x >> 5) + (threadIdx.x >> 5);
    const int tilesN = N >> 6;                 // 64-wide tiles
    const int tm = (wid / tilesN) << 4;
    const int tn = (wid % tilesN) << 6;
    if (tm >= M) return;                       // wave-uniform guard (EXEC all-1s inside)
    const int half = lane >> 4;                // selects K pair {0,1} vs {2,3}
    const int l15  = lane & 15;
    v8f acc0 = {}, acc1 = {}, acc2 = {}, acc3 = {};
    const float* arow = A + (size_t)(tm + l15) * K + (half << 1);
    const float* bptr = B + (size_t)(half << 1) * N + tn + l15;
    for (int k0 = 0; k0 < K; k0 += 4) {
        v2f a, b0, b1, b2, b3;
        a.x  = arow[0];      a.y  = arow[1];
        b0.x = bptr[0];      b0.y = bptr[N];
        b1.x = bptr[16];     b1.y = bptr[N + 16];
        b2.x = bptr[32];     b2.y = bptr[N + 32];
        b3.x = bptr[48];     b3.y = bptr[N + 48];
        acc0 = __builtin_amdgcn_wmma_f32_16x16x4_f32(false, a, false, b0, (short)0, acc0, false, false);
        acc1 = __builtin_amdgcn_wmma_f32_16x16x4_f32(false, a, false, b1, (short)0, acc1, false, false);
        acc2 = __builtin_amdgcn_wmma_f32_16x16x4_f32(false, a, false, b2, (short)0, acc2, false, false);
        acc3 = __builtin_amdgcn_wmma_f32_16x16x4_f32(false, a, false, b3, (short)0, acc3, false, false);
        arow += 4;
        bptr += (size_t)4 * N;
    }
    v8f* accs[4] = { &acc0, &acc1, &acc2, &acc3 };
#pragma unroll
    for (int t = 0; t < 4; ++t) {
        v8f& acc = *accs[t];
#pragma unroll
        for (int j = 0; j < 8; ++j) {
            const int r  = tm + j + (half << 3);
            const int cc = tn + (t << 4) + l15;
            float v = acc[j];
            if (bias) v += bias[cc];
            if (relu) v = v > 0.f ? v : 0.f;
            C[(size_t)r * N + cc] = v;
        }
    }
}

// ---------------------------------------------------------------------
__global__ void fill_f32(float* p, float v, long n) {
    long i = (long)blockIdx.x * blockDim.x + threadIdx.x;
    if (i < n) p[i] = v;
}
__global__ void fill_u32(unsigned* p, unsigned v, long n) {
    long i = (long)blockIdx.x * blockDim.x + threadIdx.x;
    if (i < n) p[i] = v;
}

// M0[d,h] = sum_c We0[d, h*32+c] * ae0[h,c]   (exact collapse of he.einsum(ae0))
__global__ void compute_M0(const float* __restrict__ We0, const float* __restrict__ ae0,
                           float* __restrict__ M0)
{
    int i = threadIdx.x;           // 0..127  -> d*4 + h
    if (i < 128) {
        int d = i >> 2, h = i & 3;
        float s = 0.f;
        for (int c = 0; c < 32; ++c) s += We0[d * 128 + h * 32 + c] * ae0[h * 32 + c];
        M0[d * 4 + h] = s;
    }
}

// per-edge: y = ReLU(edge_attr @ ep_W + ep_b); alpha[e,h] = y . M0[:,h]; also sum y
__global__ void edge_alpha_kernel(const float* __restrict__ edge_attr,
                                  const float* __restrict__ epW,
                                  const float* __restrict__ epb,
                                  const float* __restrict__ M0,
                                  float* __restrict__ alpha,
                                  float* __restrict__ ea_sum)
{
    __shared__ float sW[1024];
    __shared__ float sM[128];
    __shared__ float sB[32];
    __shared__ float sSum[32];
    for (int i = threadIdx.x; i < 1024; i += blockDim.x) sW[i] = epW[i];
    if (threadIdx.x < 128) sM[threadIdx.x] = M0[threadIdx.x];
    if (threadIdx.x < 32) { sB[threadIdx.x] = epb[threadIdx.x]; sSum[threadIdx.x] = 0.f; }
    __syncthreads();
    const long e = (long)blockIdx.x * blockDim.x + threadIdx.x;
    if (e < N_EDGES) {
        float x[32];
        const float* row = edge_attr + e * 32;
#pragma unroll
        for (int k = 0; k < 32; ++k) x[k] = row[k];
        float al0 = 0.f, al1 = 0.f, al2 = 0.f, al3 = 0.f;
        for (int j = 0; j < 32; ++j) {
            float acc = sB[j];
#pragma unroll
            for (int k = 0; k < 32; ++k) acc += x[k] * sW[k * 32 + j];
            float y = acc > 0.f ? acc : 0.f;
            al0 += y * sM[j * 4 + 0];
            al1 += y * sM[j * 4 + 1];
            al2 += y * sM[j * 4 + 2];
            al3 += y * sM[j * 4 + 3];
            atomicAdd(&sSum[j], y);
        }
        float* ap = alpha + e * 4;
        ap[0] = al0; ap[1] = al1; ap[2] = al2; ap[3] = al3;
    }
    __syncthreads();
    if (threadIdx.x < 32) atomicAdd(&ea_sum[threadIdx.x], sSum[threadIdx.x]);
}

// alpha for self-loop edges (mean edge feature . M0)
__global__ void compute_alpha_loop(const float* __restrict__ ea_sum,
                                   const float* __restrict__ M0,
                                   float* __restrict__ aloop, float invE)
{
    int h = threadIdx.x;
    if (h < 4) {
        float s = 0.f;
        for (int j = 0; j < 32; ++j) s += ea_sum[j] * invE * M0[j * 4 + h];
        aloop[h] = s;
    }
}

// ls[n,h] = hW[n,h,:] . a_src[h,:] ; ld likewise
__global__ void att_coeff(const float* __restrict__ hW, const float* __restrict__ a_src,
                          const float* __restrict__ a_dst,
                          float* __restrict__ ls, float* __restrict__ ld,
                          int n, int c)
{
    int i = blockIdx.x * blockDim.x + threadIdx.x;
    if (i >= n * HEADS) return;
    int node = i >> 2, h = i & 3;
    const float* row = hW + (size_t)node * HEADS * c + (size_t)h * c;
    float s = 0.f, d = 0.f;
    for (int k = 0; k < c; ++k) {
        s += row[k] * a_src[h * c + k];
        d += row[k] * a_dst[h * c + k];
    }
    ls[i] = s; ld[i] = d;
}

// pass 1: logits (leaky-relu) + per-(dst,head) max via ordered-uint atomicMax
__global__ void logits_pass1(const int* __restrict__ ei,
                             const float* __restrict__ ls, const float* __restrict__ ld,
                             const float* __restrict__ alpha, const float* __restrict__ aloop,
                             float* __restrict__ lgbuf, unsigned* __restrict__ segmax)
{
    int i = blockIdx.x * blockDim.x + threadIdx.x;
    if (i >= N_ETOT * 4) return;
    int e = i >> 2, h = i & 3;
    int s, d;
    if (e < N_EDGES) { s = ei[e]; d = ei[N_EDGES + e]; } else { s = d = e - N_EDGES; }
    float lg = ls[s * 4 + h] + ld[d * 4 + h];
    if (alpha) lg += (e < N_EDGES) ? alpha[(size_t)e * 4 + h] : aloop[h];
    lg = lg > 0.f ? lg : 0.2f * lg;
    lgbuf[i] = lg;
    atomicMax(&segmax[d * 4 + h], f2ord(lg));
}

// pass 2: exp(logit - segmax) in place, accumulate denominator
__global__ void logits_pass2(const int* __restrict__ ei,
                             float* __restrict__ lgbuf,
                             const unsigned* __restrict__ segmax,
                             float* __restrict__ den)
{
    int i = blockIdx.x * blockDim.x + threadIdx.x;
    if (i >= N_ETOT * 4) return;
    int e = i >> 2, h = i & 3;
    int d = (e < N_EDGES) ? ei[N_EDGES + e] : e - N_EDGES;
    float m  = ord2f(segmax[d * 4 + h]);
    float ex = __expf(lgbuf[i] - m);
    lgbuf[i] = ex;
    atomicAdd(&den[d * 4 + h], ex);
}

// pass 3: out[dst] += att * hW[src] — one block per edge, thread per 4 channels
// (float4 gather on the dominant-bandwidth path; atomics stay per-float).
__global__ void aggregate(const int* __restrict__ ei, const float* __restrict__ hW,
                          const float* __restrict__ expv, const float* __restrict__ den,
                          float* __restrict__ out, int c)
{
    int e    = blockIdx.x;
    int tid  = threadIdx.x;              // 0 .. CT/4-1
    int CT   = blockDim.x * 4;           // HEADS * c
    int base = tid * 4;                  // channel group start (same head: c % 4 == 0)
    int h    = base / c;
    int s, d;
    if (e < N_EDGES) { s = ei[e]; d = ei[N_EDGES + e]; } else { s = d = e - N_EDGES; }
    float a = expv[(size_t)e * 4 + h] / den[(size_t)d * 4 + h];
    const float4 hv = *(const float4*)(hW + (size_t)s * CT + base);
    float* op = out + (size_t)d * CT + base;
    atomicAdd(op + 0, a * hv.x);
    atomicAdd(op + 1, a * hv.y);
    atomicAdd(op + 2, a * hv.z);
    atomicAdd(op + 3, a * hv.w);
}

// bias + (optional head-mean) + residual + layernorm; writes h in place
__global__ void post_ln(const float* __restrict__ outac, const float* __restrict__ bias,
                        const float* __restrict__ g, const float* __restrict__ be,
                        float* __restrict__ h, int mean4)
{
    int n = blockIdx.x, t = threadIdx.x;          // 128 threads
    __shared__ float sm[128];
    float v;
    if (mean4) {
        float a = 0.f;
#pragma unroll
        for (int hh = 0; hh < 4; ++hh) a += outac[(size_t)n * 512 + hh * 128 + t];
        v = a * 0.25f;
    } else {
        v = outac[(size_t)n * 128 + t];
    }
    v += bias[t] + h[(size_t)n * 128 + t];
    sm[t] = v; __syncthreads();
    for (int s2 = 64; s2 > 0; s2 >>= 1) { if (t < s2) sm[t] += sm[t + s2]; __syncthreads(); }
    float mu = sm[0] * (1.f / 128.f); __syncthreads();
    float dv = v - mu;
    sm[t] = dv * dv; __syncthreads();
    for (int s2 = 64; s2 > 0; s2 >>= 1) { if (t < s2) sm[t] += sm[t + s2]; __syncthreads(); }
    float var = sm[0] * (1.f / 128.f);
    h[(size_t)n * 128 + t] = dv * rsqrtf(var + 1e-5f) * g[t] + be[t];
}

// =====================================================================
static void run_gat_layer(const float* Wl, const float* asrc, const float* adst,
                          const float* bias, const float* g, const float* be,
                          const float* alpha, const float* aloop,
                          float* h, float* hW, float* outac, float* ls, float* ld,
                          unsigned* segmax, float* den, float* lgbuf,
                          const int* ei, int CT, hipStream_t stream)
{
    const int c = CT / HEADS;
    // hW = h @ Wl  (50000x128 @ 128xCT), 16x64 tiles per wave
    int waves = (N_NODES / 16) * (CT / 64);
    gemm_wmma_f32<<<(waves + 7) / 8, 256, 0, stream>>>(h, Wl, nullptr, hW,
                                                       N_NODES, CT, 128, 0);
    att_coeff<<<(N_NODES * 4 + 255) / 256, 256, 0, stream>>>(hW, asrc, adst, ls, ld,
                                                             N_NODES, c);
    fill_u32<<<(N_NODES * 4 + 255) / 256, 256, 0, stream>>>(segmax, ORD_NEG_INF,
                                                            (long)N_NODES * 4);
    fill_f32<<<(N_NODES * 4 + 255) / 256, 256, 0, stream>>>(den, 0.f, (long)N_NODES * 4);
    long outn = (long)N_NODES * CT;
    fill_f32<<<(unsigned)((outn + 255) / 256), 256, 0, stream>>>(outac, 0.f, outn);
    int tot = N_ETOT * 4;
    logits_pass1<<<(tot + 255) / 256, 256, 0, stream>>>(ei, ls, ld, alpha, aloop,
                                                        lgbuf, segmax);
    logits_pass2<<<(tot + 255) / 256, 256, 0, stream>>>(ei, lgbuf, segmax, den);
    aggregate<<<N_ETOT, CT / 4, 0, stream>>>(ei, hW, lgbuf, den, outac, c);
    post_ln<<<N_NODES, 128, 0, stream>>>(outac, bias, g, be, h, CT == 512 ? 1 : 0);
}

extern "C" void kernel_launch(void* const* d_in, const int* in_sizes, int n_in,
                              void* d_out, int out_size, void* d_ws, size_t ws_size,
                              hipStream_t stream)
{
    const float* x     = (const float*)d_in[0];
    const int*   ei    = (const int*)  d_in[1];   // [2,E]: src then dst
    const float* eattr = (const float*)d_in[2];
    const float* W_in  = (const float*)d_in[3];
    const float* b_in  = (const float*)d_in[4];
    const float* epW   = (const float*)d_in[5];
    const float* epb   = (const float*)d_in[6];
    const float* W0    = (const float*)d_in[7];
    const float* as0   = (const float*)d_in[8];
    const float* ad0   = (const float*)d_in[9];
    const float* We0   = (const float*)d_in[10];
    const float* ae0   = (const float*)d_in[11];
    const float* bias0 = (const float*)d_in[12];
    const float* g0    = (const float*)d_in[13];
    const float* be0   = (const float*)d_in[14];
    const float* W1    = (const float*)d_in[15];
    const float* as1   = (const float*)d_in[16];
    const float* ad1   = (const float*)d_in[17];
    const float* bias1 = (const float*)d_in[18];
    const float* g1    = (const float*)d_in[19];
    const float* be1   = (const float*)d_in[20];
    const float* W2    = (const float*)d_in[21];
    const float* as2   = (const float*)d_in[22];
    const float* ad2   = (const float*)d_in[23];
    const float* bias2 = (const float*)d_in[24];
    const float* g2    = (const float*)d_in[25];
    const float* be2   = (const float*)d_in[26];
    const float* W_out = (const float*)d_in[27];
    const float* b_out = (const float*)d_in[28];

    float* ws = (float*)d_ws;
    size_t off = 0;
    float* h      = ws + off; off += (size_t)N_NODES * 128;
    float* hW     = ws + off; off += (size_t)N_NODES * 512;
    float* outac  = ws + off; off += (size_t)N_NODES * 512;
    float* ls     = ws + off; off += (size_t)N_NODES * 4;
    float* ldv    = ws + off; off += (size_t)N_NODES * 4;
    unsigned* smx = (unsigned*)(ws + off); off += (size_t)N_NODES * 4;
    float* den    = ws + off; off += (size_t)N_NODES * 4;
    float* alpha  = ws + off; off += (size_t)N_EDGES * 4;
    float* lgbuf  = ws + off; off += (size_t)N_ETOT * 4;
    float* M0     = ws + off; off += 128;
    float* easum  = ws + off; off += 32;
    float* aloop  = ws + off; off += 4;

    // input projection: h = ReLU(x @ W_in + b_in)   (50000x64 @ 64x128)
    {
        int waves = (N_NODES / 16) * (128 / 64);
        gemm_wmma_f32<<<(waves + 7) / 8, 256, 0, stream>>>(x, W_in, b_in, h,
                                                           N_NODES, 128, 64, 1);
    }
    // edge feature path (layer 0 attention bias)
    compute_M0<<<1, 128, 0, stream>>>(We0, ae0, M0);
    fill_f32<<<1, 32, 0, stream>>>(easum, 0.f, 32);
    edge_alpha_kernel<<<(N_EDGES + 255) / 256, 256, 0, stream>>>(eattr, epW, epb, M0,
                                                                 alpha, easum);
    compute_alpha_loop<<<1, 4, 0, stream>>>(easum, M0, aloop, 1.f / (float)N_EDGES);

    // GAT layers
    run_gat_layer(W0, as0, ad0, bias0, g0, be0, alpha, aloop,
                  h, hW, outac, ls, ldv, smx, den, lgbuf, ei, 128, stream);
    run_gat_layer(W1, as1, ad1, bias1, g1, be1, nullptr, nullptr,
                  h, hW, outac, ls, ldv, smx, den, lgbuf, ei, 128, stream);
    run_gat_layer(W2, as2, ad2, bias2, g2, be2, nullptr, nullptr,
                  h, hW, outac, ls, ldv, smx, den, lgbuf, ei, 512, stream);

    // output projection: d_out = h @ W_out + b_out
    {
        int waves = (N_NODES / 16) * (128 / 64);
        gemm_wmma_f32<<<(waves + 7) / 8, 256, 0, stream>>>(h, W_out, b_out,
                                                           (float*)d_out,
                                                           N_NODES, 128, 128, 0);
    }
}